// PostProcessor_59897613910738
// MI455X (gfx1250) — compile-verified
//
#include <hip/hip_runtime.h>
#include <stdint.h>

// ---------------------------------------------------------------------------
// DETR-style postprocess: score filter -> argsort -> NMS -> top-100 pack.
// B=8 images, N=4096 boxes. Three kernels:
//   1) sort_kernel : per-image bitonic sort of (score,idx) keys in LDS,
//                    emits sorted cxcywh boxes, xyxy boxes, scores, valid count
//   2) mask_kernel : suppression bitmask matrix (iou > 0.7 && j > i),
//                    LDS-tiled columns (+ gfx1250 async global->LDS b128 path)
//   3) scan_kernel : single wave32 per image; sequential greedy scan with the
//                    128-word "removed" bitmask held in registers (4 words/lane,
//                    broadcast by __shfl -- lockstep wave32, no barriers),
//                    emits first 100 kept boxes + count, zero-pads the rest.
// ---------------------------------------------------------------------------

#define SCORE_T 0.5f
#define NMS_T   0.7f
#define MAX_DET 100
#define BB      8
#define NBOX    4096
#define NW      (NBOX / 32)   // 128 mask words per row
#define TILE    128           // column tile for IoU pass

#if defined(__gfx1250__) && __has_builtin(__builtin_amdgcn_global_load_async_to_lds_b128) && __has_builtin(__builtin_amdgcn_s_wait_asynccnt)
#define USE_ASYNC_LDS 1
#else
#define USE_ASYNC_LDS 0
#endif

#if USE_ASYNC_LDS
// Builtin signature (from clang diagnostic): first param is
// '<4 x i32> addrspace(1)*' (printed as "int vector __device__ *"), second is
// the LDS-side pointer in addrspace(3).
typedef int v4i_t __attribute__((vector_size(4 * sizeof(int))));
typedef __attribute__((address_space(1))) v4i_t as1_v4i;
typedef __attribute__((address_space(3))) v4i_t as3_v4i;
#endif

// ------------------------------ kernel 1 -----------------------------------
// One block per image. Key = (orderable_score_bits << 32) | (N-1-idx)
// sorted descending => score desc, ties by index asc (matches stable argsort).
__global__ __launch_bounds__(512)
void sort_kernel(const float* __restrict__ boxes,   // [B,N,4] cxcywh
                 const float* __restrict__ scores,  // [B,N]
                 float* __restrict__ sBox,          // [B,N,4] sorted cxcywh
                 float* __restrict__ sXYXY,         // [B,N,4] sorted xyxy
                 float* __restrict__ sScore,        // [B,N]   sorted scores
                 int*   __restrict__ vCount) {      // [B]
    __shared__ unsigned long long sk[NBOX];
    __shared__ int cnt;
    const int b   = blockIdx.x;
    const int tid = threadIdx.x;
    if (tid == 0) cnt = 0;
    __syncthreads();

    int local = 0;
    for (int i = tid; i < NBOX; i += 512) {
        float s = scores[(size_t)b * NBOX + i];
        bool valid = s > SCORE_T;
        local += valid ? 1 : 0;
        unsigned bits = __float_as_uint(valid ? s : -__builtin_inff());
        // monotone float->uint mapping (handles -inf correctly)
        unsigned u = bits ^ ((unsigned)((int)bits >> 31) | 0x80000000u);
        sk[i] = ((unsigned long long)u << 32) | (unsigned)(NBOX - 1 - i);
    }
    atomicAdd(&cnt, local);

    // Bitonic sort, descending.
    for (int k = 2; k <= NBOX; k <<= 1) {
        for (int j = k >> 1; j > 0; j >>= 1) {
            __syncthreads();
            for (int i = tid; i < NBOX; i += 512) {
                int ixj = i ^ j;
                if (ixj > i) {
                    unsigned long long a = sk[i], c = sk[ixj];
                    bool up = (i & k) == 0;               // descending region
                    if (up ? (a < c) : (a > c)) { sk[i] = c; sk[ixj] = a; }
                }
            }
        }
    }
    __syncthreads();

    for (int r = tid; r < NBOX; r += 512) {
        unsigned idx = NBOX - 1u - (unsigned)(sk[r] & 0xFFFFFFFFu);
        const float* src = boxes + ((size_t)b * NBOX + idx) * 4;
        float cx = src[0], cy = src[1], w = src[2], h = src[3];
        size_t o = ((size_t)b * NBOX + r) * 4;
        sBox[o + 0] = cx; sBox[o + 1] = cy; sBox[o + 2] = w; sBox[o + 3] = h;
        sXYXY[o + 0] = cx - 0.5f * w;
        sXYXY[o + 1] = cy - 0.5f * h;
        sXYXY[o + 2] = cx + 0.5f * w;
        sXYXY[o + 3] = cy + 0.5f * h;
        sScore[(size_t)b * NBOX + r] = scores[(size_t)b * NBOX + idx];
    }
    if (tid == 0) vCount[b] = cnt;
}

// ------------------------------ kernel 2 -----------------------------------
// grid = (N/256, B), block = 256. Thread t owns row i = blockIdx.x*256 + t.
// Columns staged in LDS tiles of 128 boxes; bit set iff iou>0.7 and j>i.
__global__ __launch_bounds__(256)
void mask_kernel(const float* __restrict__ sXYXY,
                 const int*   __restrict__ vCount,
                 unsigned*    __restrict__ masks) {   // [B,N,NW]
    const int b = blockIdx.y;
    const int V = vCount[b];
    if ((int)(blockIdx.x * 256) >= V) return;         // uniform per block
    const int i = blockIdx.x * 256 + threadIdx.x;

    __shared__ __align__(16) float tb[TILE * 4];
    __shared__ float tarea[TILE];

    const float* base = sXYXY + (size_t)b * NBOX * 4;
    float ax1 = 0.f, ay1 = 0.f, ax2 = 0.f, ay2 = 0.f, aarea = 0.f;
    const bool rowOn = (i < V);
    if (rowOn) {
        const float* p = base + (size_t)i * 4;
        ax1 = p[0]; ay1 = p[1]; ax2 = p[2]; ay2 = p[3];
        aarea = fmaxf(ax2 - ax1, 0.f) * fmaxf(ay2 - ay1, 0.f);
    }
    unsigned* myrow = masks + ((size_t)b * NBOX + i) * NW;

    for (int j0 = 0; j0 < NBOX; j0 += TILE) {
        __syncthreads();
        // prefetch next column tile into cache (gfx1250: global_prefetch_b8)
        if (threadIdx.x < TILE && j0 + TILE < NBOX)
            __builtin_prefetch(base + (size_t)(j0 + TILE + threadIdx.x) * 4, 0, 0);

#if USE_ASYNC_LDS
        if (threadIdx.x < TILE) {
            const float* src = base + (size_t)(j0 + threadIdx.x) * 4;
            __builtin_amdgcn_global_load_async_to_lds_b128(
                (as1_v4i*)(uintptr_t)src,
                (as3_v4i*)(unsigned)(uintptr_t)(&tb[threadIdx.x * 4]),
                0, 0);
            __builtin_amdgcn_s_wait_asynccnt(0);
        }
#else
        for (int t = threadIdx.x; t < TILE * 4; t += 256)
            tb[t] = base[(size_t)j0 * 4 + t];
#endif
        __syncthreads();
        if (threadIdx.x < TILE) {
            float x1 = tb[threadIdx.x * 4 + 0], y1 = tb[threadIdx.x * 4 + 1];
            float x2 = tb[threadIdx.x * 4 + 2], y2 = tb[threadIdx.x * 4 + 3];
            tarea[threadIdx.x] = fmaxf(x2 - x1, 0.f) * fmaxf(y2 - y1, 0.f);
        }
        __syncthreads();

        if (rowOn) {
            for (int w = 0; w < TILE / 32; ++w) {
                unsigned bits = 0u;
                for (int s2 = 0; s2 < 32; ++s2) {
                    int jl = w * 32 + s2;
                    int j  = j0 + jl;
                    float ix1 = fmaxf(ax1, tb[jl * 4 + 0]);
                    float iy1 = fmaxf(ay1, tb[jl * 4 + 1]);
                    float ix2 = fminf(ax2, tb[jl * 4 + 2]);
                    float iy2 = fminf(ay2, tb[jl * 4 + 3]);
                    float iw = fmaxf(ix2 - ix1, 0.f);
                    float ih = fmaxf(iy2 - iy1, 0.f);
                    float inter = iw * ih;
                    float uni = fmaxf(aarea + tarea[jl] - inter, 1e-9f);
                    // iou > T  <=>  inter > T*uni  (division-free)
                    bits |= ((inter > NMS_T * uni) && (j > i)) ? (1u << s2) : 0u;
                }
                myrow[(j0 >> 5) + w] = bits;
            }
        }
    }
}

// ------------------------------ kernel 3 -----------------------------------
// One wave32 per image. Lane l holds removed words {l, l+32, l+64, l+96}.
// Lockstep wave => no barriers; word broadcast via __shfl.
__global__ __launch_bounds__(32)
void scan_kernel(const float* __restrict__ sBox,
                 const float* __restrict__ sScore,
                 const int*   __restrict__ vCount,
                 const unsigned* __restrict__ masks,
                 float* __restrict__ outPacked,   // [B,MAX_DET,5]
                 int*   __restrict__ outNum) {    // [B]
    const int b    = blockIdx.x;
    const int lane = threadIdx.x;
    unsigned rm0 = 0u, rm1 = 0u, rm2 = 0u, rm3 = 0u;
    int V = vCount[b];
    if (V > NBOX) V = NBOX;
    const unsigned* mb = masks + (size_t)b * NBOX * NW;
    float* op = outPacked + (size_t)b * MAX_DET * 5;
    int outCount = 0;

    for (int i = 0; i < V; ++i) {
        int wi = i >> 5;                 // removed word index (uniform)
        int q  = wi >> 5;                // which register bank
        unsigned word = (q == 0) ? rm0 : (q == 1) ? rm1 : (q == 2) ? rm2 : rm3;
        word = __shfl(word, wi & 31, 32);
        if (((word >> (i & 31)) & 1u) == 0u) {     // box i survives
            const unsigned* row = mb + (size_t)i * NW;
            rm0 |= row[lane];
            rm1 |= row[lane + 32];
            rm2 |= row[lane + 64];
            rm3 |= row[lane + 96];
            if (lane < 4)
                op[(size_t)outCount * 5 + lane] = sBox[((size_t)b * NBOX + i) * 4 + lane];
            else if (lane == 4)
                op[(size_t)outCount * 5 + 4] = sScore[(size_t)b * NBOX + i];
            ++outCount;
            if (outCount == MAX_DET) break;        // num caps at 100 anyway
        }
    }
    // zero-pad remaining detection slots
    for (int t = outCount * 5 + lane; t < MAX_DET * 5; t += 32) op[t] = 0.f;
    if (lane == 0) outNum[b] = outCount;           // = min(total kept, 100)
}

// ------------------------------ launcher -----------------------------------
extern "C" void kernel_launch(void* const* d_in, const int* in_sizes, int n_in,
                              void* d_out, int out_size, void* d_ws, size_t ws_size,
                              hipStream_t stream) {
    const float* boxes  = (const float*)d_in[0];   // [8,4096,4] cxcywh
    const float* scores = (const float*)d_in[1];   // [8,4096]

    char* ws = (char*)d_ws;
    size_t o = 0;
    float* sBox   = (float*)(ws + o); o += (size_t)BB * NBOX * 4 * sizeof(float); // 512 KB
    float* sXYXY  = (float*)(ws + o); o += (size_t)BB * NBOX * 4 * sizeof(float); // 512 KB
    float* sScore = (float*)(ws + o); o += (size_t)BB * NBOX * sizeof(float);     // 128 KB
    int*   vCnt   = (int*)(ws + o);   o += 256;
    unsigned* masks = (unsigned*)(ws + o);          // 8*4096*128*4 = 16 MB

    float* outPacked = (float*)d_out;               // [8,100,5] floats
    int*   outNum    = (int*)((char*)d_out + (size_t)BB * MAX_DET * 5 * sizeof(float));

    sort_kernel<<<BB, 512, 0, stream>>>(boxes, scores, sBox, sXYXY, sScore, vCnt);
    mask_kernel<<<dim3(NBOX / 256, BB), 256, 0, stream>>>(sXYXY, vCnt, masks);
    scan_kernel<<<BB, 32, 0, stream>>>(sBox, sScore, vCnt, masks, outPacked, outNum);
}